// ReinhardNormalRGB_30142080483896
// MI455X (gfx1250) — compile-verified
//
#include <hip/hip_runtime.h>

typedef float v2f __attribute__((ext_vector_type(2)));
typedef float v8f __attribute__((ext_vector_type(8)));

#define LAB_EPS 0.008856f
#define LAB_KAP 7.787f

// ---------- fast transcendentals (v_log_f32 / v_exp_f32) ----------
__device__ __forceinline__ float fpow(float x, float e) {
    // x^e = exp2(e * log2(x)); v_log_f32 is log2, v_exp_f32 is exp2
    return __builtin_amdgcn_exp2f(__builtin_amdgcn_logf(x) * e);
}

__device__ __forceinline__ float srgb_to_lin(float c) {
    float t = (c + 0.055f) * (1.0f / 1.055f);
    float p = fpow(t, 2.4f);
    return c <= 0.04045f ? c * (1.0f / 12.92f) : p;
}

__device__ __forceinline__ float lab_f(float x) {
    float cb = fpow(x, 1.0f / 3.0f);                 // cbrt for x > eps
    return x > LAB_EPS ? cb : LAB_KAP * x + 16.0f / 116.0f;
}

__device__ __forceinline__ float lab_finv(float f) {
    float f3 = f * f * f;
    return f3 > LAB_EPS ? f3 : (f - 16.0f / 116.0f) * (1.0f / LAB_KAP);
}

__device__ __forceinline__ float lin_to_srgb(float c) {
    c = fminf(fmaxf(c, 0.0f), 1.0f);
    float s = 1.055f * fpow(c, 1.0f / 2.4f) - 0.055f;
    return c <= 0.0031308f ? 12.92f * c : s;
}

// ---------- cross-half lane swap: single ds_swizzle, no exec games ----------
// bit-mode: new_lane = ((lane & 0x1F) | 0x00) ^ 0x10  -> SWAPX16
__device__ __forceinline__ float swap16(float x) {
    return __int_as_float(
        __builtin_amdgcn_ds_swizzle(__float_as_int(x), 0x401F));
}

// ---------- WMMA helpers (V_WMMA_F32_16X16X4_F32, wave32) ----------
// A operand layout (ISA 7.12.2, 32-bit A 16x4): lanes 0-15 hold row M=lane,
// VGPR0=K0 / VGPR1=K1; lanes 16-31 hold row M=lane-16, VGPR0=K2 / VGPR1=K3.
// BASE selects which D rows (0..2 or 8..10) the 3x3 matrix occupies.
template <unsigned BASE>
__device__ __forceinline__ v2f wmma_matA(unsigned lane,
        float m00, float m01, float m02,
        float m10, float m11, float m12,
        float m20, float m21, float m22) {
    unsigned r = lane & 15u;
    bool lo = lane < 16u;
    float c0 = (r == BASE + 0) ? m00 : (r == BASE + 1) ? m10 : (r == BASE + 2) ? m20 : 0.0f;
    float c1 = (r == BASE + 0) ? m01 : (r == BASE + 1) ? m11 : (r == BASE + 2) ? m21 : 0.0f;
    float c2 = (r == BASE + 0) ? m02 : (r == BASE + 1) ? m12 : (r == BASE + 2) ? m22 : 0.0f;
    v2f a;
    a.x = lo ? c0 : c2;   // {K0 | K2}
    a.y = lo ? c1 : 0.0f; // {K1 | K3(pad)}
    return a;
}

// D[m][n] = sum_k A[m][k] * B[k][n];  B columns = pixels.
// Full-wave SoA in/out (lane i <-> pixel i, 3 regs).
// B layout: VGPR0 = {K0 lanes0-15 | K2 lanes16-31}, VGPR1 = {K1 | K3(pad)}.
// D layout: VGPR j = {row M=j lanes0-15 | row M=j+8 lanes16-31}.
// Alo has the matrix at rows 0-2 (results for px 0-15 land in lanes 0-15);
// Ahi has it at rows 8-10 (results for px 16-31 land in lanes 16-31) -> the
// recombine is a plain cndmask, no cross-lane op on the outputs.
__device__ __forceinline__ void mat3_wmma(bool lo, v2f Alo, v2f Ahi,
        float c0, float c1, float c2,
        float& o0, float& o1, float& o2) {
    float s0 = swap16(c0), s1 = swap16(c1), s2 = swap16(c2);
    v2f bA, bB;
    bA.x = lo ? c0 : s2;  bA.y = lo ? c1 : 0.0f;   // pixels  0..15
    bB.x = lo ? s0 : c2;  bB.y = lo ? s1 : 0.0f;   // pixels 16..31
    v8f z = {};
    v8f dA = __builtin_amdgcn_wmma_f32_16x16x4_f32(false, Alo, false, bA,
                                                   (short)0, z, false, false);
    v8f dB = __builtin_amdgcn_wmma_f32_16x16x4_f32(false, Ahi, false, bB,
                                                   (short)0, z, false, false);
    o0 = lo ? dA[0] : dB[0];
    o1 = lo ? dA[1] : dB[1];
    o2 = lo ? dA[2] : dB[2];
}

// ---------- constants: white fold into the two matrices ----------
#define WX 0.950456f
#define WZ 1.088754f

// RGB2XYZ rows pre-divided by WHITE
#define A1ARGS \
    0.412453f / WX, 0.357580f / WX, 0.180423f / WX, \
    0.212671f,      0.715160f,      0.072169f,      \
    0.019334f / WZ, 0.119193f / WZ, 0.950227f / WZ
// XYZ2RGB columns pre-multiplied by WHITE
#define A2ARGS \
     3.240479f * WX, -1.537150f, -0.498535f * WZ, \
    -0.969256f * WX,  1.875992f,  0.041556f * WZ, \
     0.055648f * WX, -0.204043f,  1.057311f * WZ

__global__ void __launch_bounds__(256)
color_transfer_kernel(const float* __restrict__ img,
                      const float* __restrict__ sm, const float* __restrict__ ss,
                      const float* __restrict__ dm, const float* __restrict__ dsd,
                      const int*  __restrict__ mask,
                      float* __restrict__ out, long long pxPerImg) {
    const int b = blockIdx.y;
    const long long imgBase = (long long)b * pxPerImg * 3LL;
    const int startPx = blockIdx.x * 1024;
    const int t = threadIdx.x;
    const float* src = img + imgBase;
    float* dst = out + imgBase;

    if (mask[b] == 0) {
        // identity: vectorized streaming copy (1024 px = 768 float4 / block)
        const float4* s4 = (const float4*)(src + (long long)startPx * 3);
        float4* d4 = (float4*)(dst + (long long)startPx * 3);
        #pragma unroll
        for (int k = 0; k < 3; ++k)
            d4[t + k * 256] = s4[t + k * 256];
        return;
    }

    // Fold the LAB-space affine remap into an affine map on (fx,fy,fz):
    //   fx' = sa*fx + (sL-sa)*fy + kx
    //   fy' = sL*fy + ky
    //   fz' = sb*fz + (sL-sb)*fy + kz
    float sL = dsd[0] / ss[0], sa = dsd[1] / ss[1], sb = dsd[2] / ss[2];
    float oL = dm[0] - sm[0] * sL;
    float oa = dm[1] - sm[1] * sa;
    float ob = dm[2] - sm[2] * sb;
    float ky = (oL + 16.0f - 16.0f * sL) * (1.0f / 116.0f);
    float kx = ky + oa * (1.0f / 500.0f);
    float kz = ky - ob * (1.0f / 200.0f);

    const unsigned lane = (unsigned)t & 31u;
    const bool lo = lane < 16u;

    v2f A1lo = wmma_matA<0>(lane, A1ARGS);
    v2f A1hi = wmma_matA<8>(lane, A1ARGS);
    v2f A2lo = wmma_matA<0>(lane, A2ARGS);
    v2f A2hi = wmma_matA<8>(lane, A2ARGS);

    #pragma unroll 2
    for (int k = 0; k < 4; ++k) {
        const long long p = (long long)startPx + k * 256 + t;
        const float* pp = src + p * 3LL;
        float r = pp[0], g = pp[1], bch = pp[2];

        // sRGB -> linear (per channel, full wave)
        float lr = srgb_to_lin(r);
        float lg = srgb_to_lin(g);
        float lb = srgb_to_lin(bch);

        // linear RGB -> XYZ/white via matrix pipe
        float X, Y, Z;
        mat3_wmma(lo, A1lo, A1hi, lr, lg, lb, X, Y, Z);

        // LAB forward nonlinearity
        float fx = lab_f(X), fy = lab_f(Y), fz = lab_f(Z);

        // fused normalize(src) -> denormalize(dst) in f-space
        float fx2 = sa * fx + (sL - sa) * fy + kx;
        float fy2 = sL * fy + ky;
        float fz2 = sb * fz + (sL - sb) * fy + kz;

        // LAB inverse nonlinearity
        float Xp = lab_finv(fx2), Yp = lab_finv(fy2), Zp = lab_finv(fz2);

        // XYZ*white -> linear RGB via matrix pipe
        float R, G, B;
        mat3_wmma(lo, A2lo, A2hi, Xp, Yp, Zp, R, G, B);

        float* q = dst + p * 3LL;
        q[0] = lin_to_srgb(R);
        q[1] = lin_to_srgb(G);
        q[2] = lin_to_srgb(B);
    }
}

extern "C" void kernel_launch(void* const* d_in, const int* in_sizes, int n_in,
                              void* d_out, int out_size, void* d_ws, size_t ws_size,
                              hipStream_t stream) {
    const float* img = (const float*)d_in[0];
    const float* sm  = (const float*)d_in[1];
    const float* ss  = (const float*)d_in[2];
    const float* dm  = (const float*)d_in[3];
    const float* dsd = (const float*)d_in[4];
    const int*  mask = (const int*)d_in[5];
    float* out = (float*)d_out;

    const int B = in_sizes[5];                               // 32 images
    const long long pxPerImg = (long long)in_sizes[0] / (3LL * B); // 262144
    const int blocksX = (int)(pxPerImg / 1024);              // 256

    color_transfer_kernel<<<dim3(blocksX, B), 256, 0, stream>>>(
        img, sm, ss, dm, dsd, mask, out, pxPerImg);
}